// GCN_59261958750657
// MI455X (gfx1250) — compile-verified
//
#include <hip/hip_runtime.h>

typedef __attribute__((ext_vector_type(2))) float v2f;
typedef __attribute__((ext_vector_type(8))) float v8f;

#define NNODE 50000
#define NEDGE 1600000
#define HDIM  128
#define NH    (NNODE * HDIM)

// Async global->LDS single-dword copy (gfx1250, tracked by ASYNCcnt).
// Per-lane LDS dest offset and per-lane global src address -> ideal for
// permuting weights into WMMA fragment layout while staging into LDS.
__device__ __forceinline__ void async_g2l_b32(unsigned lds_off, const void* gaddr) {
  asm volatile("global_load_async_to_lds_b32 %0, %1, off"
               :: "v"(lds_off), "v"(gaddr)
               : "memory");
}
__device__ __forceinline__ void wait_async0() {
  asm volatile("s_wait_asynccnt 0x0" ::: "memory");
}

// ---------------------------------------------------------------------------
// WMMA fp32 GEMM: C[M x N] = act(A[M x KD] @ W[KD x N] + bias)
// 256 threads = 8 waves per block; W is staged once per block into LDS in
// packed B-fragment-pair layout (v2f per lane per (kchunk,ntile)), so the
// inner loop does one ds_load_b64 per fragment. Each wave computes 2 M-tiles
// (32 rows) so each B fragment feeds two v_wmma_f32_16x16x4_f32.
// Fragment layouts (wave32):
//   A 16x4:  lane l -> row = l&15, K = (l>>4)*2 .. +1
//   B 4x16:  lane l -> col = l&15, K = (l>>4)*2 .. +1
//   C 16x16: lane l, vgpr v -> row = v + 8*(l>>4), col = l&15
// ---------------------------------------------------------------------------
template <int KD, int NT>
__global__ __launch_bounds__(256) void wmma_gemm_bias_act(
    const float* __restrict__ A, const float* __restrict__ W,
    const float* __restrict__ bias, float* __restrict__ C,
    int do_relu, int mtiles) {
  constexpr int N     = NT * 16;
  constexpr int CH    = KD / 4;          // K-chunks of 4
  constexpr int PAIRS = CH * NT * 32;    // fragment pairs in LDS
  __shared__ v2f sw[PAIRS];

  const int tid = threadIdx.x;

  // ---- stage W into LDS, permuted into fragment-pair layout (async) ----
  const unsigned lbase = (unsigned)(size_t)&sw[0];
  for (int i = tid; i < PAIRS; i += 256) {
    const int c   = i / (NT * 32);
    const int r   = i - c * (NT * 32);
    const int nt  = r >> 5;
    const int ln  = r & 31;
    const int ka  = 4 * c + 2 * (ln >> 4);
    const int col = nt * 16 + (ln & 15);
    const float* g = W + (size_t)ka * N + col;
    async_g2l_b32(lbase + (unsigned)i * 8u,      g);       // K = ka
    async_g2l_b32(lbase + (unsigned)i * 8u + 4u, g + N);   // K = ka+1
  }
  wait_async0();
  __syncthreads();

  const int lane  = tid & 31;
  const int wave  = tid >> 5;
  const int r16   = lane & 15;
  const int khalf = lane >> 4;

  const int t0 = (blockIdx.x * 8 + wave) * 2;
  if (t0 >= mtiles) return;              // wave-uniform exit (after barrier)
  const int t1   = t0 + 1;
  const bool has1 = (t1 < mtiles);

  const float* A0 = A + (size_t)(t0 * 16 + r16) * KD;
  const float* A1 = A + (size_t)((has1 ? t1 : t0) * 16 + r16) * KD;

  v8f acc0[NT] = {};
  v8f acc1[NT] = {};

  for (int c = 0; c < CH; ++c) {
    const int ka = 4 * c + 2 * khalf;
    const v2f a0 = *(const v2f*)(A0 + ka);
    const v2f a1 = *(const v2f*)(A1 + ka);
    const v2f* brow = &sw[(size_t)c * NT * 32 + lane];
#pragma unroll
    for (int nt = 0; nt < NT; ++nt) {
      const v2f b = brow[nt * 32];                       // ds_load_b64
      acc0[nt] = __builtin_amdgcn_wmma_f32_16x16x4_f32(
          false, a0, false, b, (short)0, acc0[nt], false, false);
      acc1[nt] = __builtin_amdgcn_wmma_f32_16x16x4_f32(
          false, a1, false, b, (short)0, acc1[nt], false, false);
    }
  }

#pragma unroll
  for (int nt = 0; nt < NT; ++nt) {
    const int col = nt * 16 + r16;
    const float bv = bias[col];
#pragma unroll
    for (int v = 0; v < 8; ++v) {
      const int rr = v + 8 * khalf;
      float x0 = acc0[nt][v] + bv;
      if (do_relu) x0 = fmaxf(x0, 0.0f);
      C[(size_t)(t0 * 16 + rr) * N + col] = x0;
      if (has1) {
        float x1 = acc1[nt][v] + bv;
        if (do_relu) x1 = fmaxf(x1, 0.0f);
        C[(size_t)(t1 * 16 + rr) * N + col] = x1;
      }
    }
  }
}

// ---------------------------------------------------------------------------
// Fused GINE edge kernel (4 waves/block, 1 edge-tile of 16 edges per wave):
//   e = edge_attr[16x16] @ W_edge[16x128] + b_edge   (WMMA fp32, We in LDS)
//   msg = relu(h[src] + e);  atomicAdd(agg[dst], msg)
// agg must be pre-initialized to h (z = h + segment_sum(msg)).
// ---------------------------------------------------------------------------
__global__ __launch_bounds__(128) void gine_edge_kernel(
    const float* __restrict__ EA,   // [E,16]
    const float* __restrict__ We,   // [16,128]
    const float* __restrict__ be,   // [128]
    const int* __restrict__ src, const int* __restrict__ dst,
    const float* __restrict__ h, float* __restrict__ agg) {
  constexpr int NT = 8, CH = 4;          // K=16
  constexpr int PAIRS = CH * NT * 32;    // 1024 pairs = 8KB
  __shared__ v2f sw[PAIRS];

  const int tid = threadIdx.x;
  const unsigned lbase = (unsigned)(size_t)&sw[0];
  for (int i = tid; i < PAIRS; i += 128) {
    const int c   = i / (NT * 32);
    const int r   = i - c * (NT * 32);
    const int nt  = r >> 5;
    const int ln  = r & 31;
    const int ka  = 4 * c + 2 * (ln >> 4);
    const int col = nt * 16 + (ln & 15);
    const float* g = We + (size_t)ka * 128 + col;
    async_g2l_b32(lbase + (unsigned)i * 8u,      g);
    async_g2l_b32(lbase + (unsigned)i * 8u + 4u, g + 128);
  }
  wait_async0();
  __syncthreads();

  const int lane  = tid & 31;
  const int wave  = tid >> 5;
  const int r16   = lane & 15;
  const int khalf = lane >> 4;
  const int ebase = (blockIdx.x * 4 + wave) * 16;   // E = 1.6M exact

  const float* Arow = EA + (size_t)(ebase + r16) * 16;

  v8f acc[NT] = {};
#pragma unroll
  for (int c = 0; c < CH; ++c) {
    const int ka = 4 * c + 2 * khalf;
    const v2f a = *(const v2f*)(Arow + ka);
    const v2f* brow = &sw[(size_t)c * NT * 32 + lane];
#pragma unroll
    for (int nt = 0; nt < NT; ++nt) {
      const v2f b = brow[nt * 32];
      acc[nt] = __builtin_amdgcn_wmma_f32_16x16x4_f32(
          false, a, false, b, (short)0, acc[nt], false, false);
    }
  }

  int sidx[8], didx[8];
#pragma unroll
  for (int v = 0; v < 8; ++v) {
    const int m = v + 8 * khalf;     // edge row this lane holds in C vgpr v
    sidx[v] = src[ebase + m];
    didx[v] = dst[ebase + m];
  }

#pragma unroll
  for (int nt = 0; nt < NT; ++nt) {
    const int col = nt * 16 + r16;
    const float bv = be[col];
#pragma unroll
    for (int v = 0; v < 8; ++v) {
      float val = acc[nt][v] + bv + h[(size_t)sidx[v] * 128 + col];
      val = fmaxf(val, 0.0f);
      atomicAdd(&agg[(size_t)didx[v] * 128 + col], val);
    }
  }
}

// z = h  (float4 copy)
__global__ void copy_f4(float4* __restrict__ d, const float4* __restrict__ s, int n4) {
  int i = blockIdx.x * blockDim.x + threadIdx.x;
  if (i < n4) d[i] = s[i];
}

// Final head: out[r,0:3] = t2[r,0:64] @ W_l3[64,3] + b_l3
__global__ void head3_kernel(const float* __restrict__ t2,
                             const float* __restrict__ W,
                             const float* __restrict__ b,
                             float* __restrict__ out, int n) {
  int r = blockIdx.x * blockDim.x + threadIdx.x;
  if (r >= n) return;
  const float* row = t2 + (size_t)r * 64;
  float o0 = b[0], o1 = b[1], o2 = b[2];
#pragma unroll 4
  for (int k = 0; k < 64; ++k) {
    const float v = row[k];
    o0 += v * W[k * 3 + 0];
    o1 += v * W[k * 3 + 1];
    o2 += v * W[k * 3 + 2];
  }
  out[(size_t)r * 3 + 0] = o0;
  out[(size_t)r * 3 + 1] = o1;
  out[(size_t)r * 3 + 2] = o2;
}

extern "C" void kernel_launch(void* const* d_in, const int* in_sizes, int n_in,
                              void* d_out, int out_size, void* d_ws, size_t ws_size,
                              hipStream_t stream) {
  (void)in_sizes; (void)n_in; (void)out_size; (void)ws_size;

  const float* x       = (const float*)d_in[0];
  const int*   eidx    = (const int*)  d_in[1];
  const float* eattr   = (const float*)d_in[2];
  const float* W_first = (const float*)d_in[3];
  const float* b_first = (const float*)d_in[4];
  const float* W_edge  = (const float*)d_in[5];
  const float* b_edge  = (const float*)d_in[6];
  const float* W1      = (const float*)d_in[7];
  const float* b1      = (const float*)d_in[8];
  const float* W2      = (const float*)d_in[9];
  const float* b2      = (const float*)d_in[10];
  const float* W_l1    = (const float*)d_in[11];
  const float* b_l1    = (const float*)d_in[12];
  const float* W_l2    = (const float*)d_in[13];
  const float* b_l2    = (const float*)d_in[14];
  const float* W_l3    = (const float*)d_in[15];
  const float* b_l3    = (const float*)d_in[16];

  const int* src = eidx;
  const int* dst = eidx + NEDGE;

  float* h   = (float*)d_ws;
  float* z   = h + NH;
  float* tmp = z + NH;

  const int MT = NNODE / 16;                 // 3125 M-tiles, exact
  const int GEMM_BLOCKS = (MT + 15) / 16;    // 8 waves * 2 tiles per block
  const int EDGE_BLOCKS = NEDGE / 16 / 4;    // 25000, exact

  // h = relu(x @ W_first + b_first)
  wmma_gemm_bias_act<64, 8><<<GEMM_BLOCKS, 256, 0, stream>>>(x, W_first, b_first, h, 1, MT);

  for (int l = 0; l < 6; ++l) {
    copy_f4<<<(NH / 4 + 255) / 256, 256, 0, stream>>>((float4*)z, (const float4*)h, NH / 4);
    gine_edge_kernel<<<EDGE_BLOCKS, 128, 0, stream>>>(eattr,
                                                      W_edge + (size_t)l * 16 * 128,
                                                      b_edge + (size_t)l * 128,
                                                      src, dst, h, z);
    wmma_gemm_bias_act<128, 8><<<GEMM_BLOCKS, 256, 0, stream>>>(z,   W1 + (size_t)l * 128 * 128,
                                                                b1 + (size_t)l * 128, tmp, 1, MT);
    wmma_gemm_bias_act<128, 8><<<GEMM_BLOCKS, 256, 0, stream>>>(tmp, W2 + (size_t)l * 128 * 128,
                                                                b2 + (size_t)l * 128, h, 1, MT);
  }

  wmma_gemm_bias_act<128, 8><<<GEMM_BLOCKS, 256, 0, stream>>>(h, W_l1, b_l1, z, 1, MT);
  wmma_gemm_bias_act<128, 4><<<GEMM_BLOCKS, 256, 0, stream>>>(z, W_l2, b_l2, tmp, 1, MT);
  head3_kernel<<<(NNODE + 255) / 256, 256, 0, stream>>>(tmp, W_l3, b_l3, (float*)d_out, NNODE);
}